// SAGE_44487271252165
// MI455X (gfx1250) — compile-verified
//
#include <hip/hip_runtime.h>
#include <hip/hip_bf16.h>

#define NN   100000
#define EDG  1600000
#define DH   128
#define DOUTC 32
#define BN_EPS 1e-5f

typedef __attribute__((ext_vector_type(2))) float v2f;
typedef __attribute__((ext_vector_type(8))) float v8f;

// ---------------------------------------------------------------- utilities
__global__ void zero_kernel(float* __restrict__ p, int n) {
    int i = blockIdx.x * blockDim.x + threadIdx.x;
    if (i < n) p[i] = 0.0f;
}

__global__ void deg_count_kernel(const int* __restrict__ dst, float* __restrict__ deg, int e_cnt) {
    int e = blockIdx.x * blockDim.x + threadIdx.x;
    if (e < e_cnt) atomicAdd(&deg[dst[e]], 1.0f);
}

__global__ void inv_deg_kernel(float* __restrict__ deg, int n) {
    int i = blockIdx.x * blockDim.x + threadIdx.x;
    if (i < n) deg[i] = 1.0f / fmaxf(deg[i], 1.0f);
}

// ------------------------------------------------- edge scatter (segment sum)
// One thread handles one (edge, 4-feature chunk): float4 gather (L2-resident
// feature matrix, 51MB < 192MB L2) + 4 f32 atomics into the segment sum.
__global__ void scatter_kernel(const int* __restrict__ src, const int* __restrict__ dst,
                               const float* __restrict__ x, float* __restrict__ s, int e_cnt) {
    int t = blockIdx.x * blockDim.x + threadIdx.x;
    int e = t >> 5;                    // 32 chunks of 4 floats = 128 features
    if (e >= e_cnt) return;
    int c  = (t & 31) << 2;
    int si = src[e] * DH + c;
    int di = dst[e] * DH + c;
    float4 v = *reinterpret_cast<const float4*>(x + si);
    atomicAdd(s + di + 0, v.x);
    atomicAdd(s + di + 1, v.y);
    atomicAdd(s + di + 2, v.z);
    atomicAdd(s + di + 3, v.w);
}

// --------------------------------------------------------- fused WMMA GEMM
// D[16 x NT*16 stripe] = (Am * inv_deg) @ Wm + Ax @ Wx + bias
// fp32 WMMA 16x16x4; optional column sum / sum-of-squares accumulation for BN.
// One wave per block; grid.x = NN/16 (N is an exact multiple of 16 so EXEC
// stays all-ones as WMMA requires).
template <int NT>
__global__ void __launch_bounds__(32)
gemm_bn_kernel(const float* __restrict__ Am, const float* __restrict__ invdeg,
               const float* __restrict__ Ax,
               const float* __restrict__ Wm, const float* __restrict__ Wx,
               const float* __restrict__ bias,
               float* __restrict__ D,
               float* __restrict__ colsum, float* __restrict__ colsumsq,
               int K) {
    const int lane = threadIdx.x;
    const int half = lane >> 4;            // which 16-lane half
    const int l16  = lane & 15;
    const int row  = blockIdx.x * 16 + l16;  // A-matrix row owned by this lane
    const int ncol = NT * 16;

    v8f acc[NT];
#pragma unroll
    for (int t = 0; t < NT; ++t) acc[t] = (v8f)0.0f;

    // ---- path 1: mean-aggregated neighbor features (scaled by 1/deg) ----
    if (Am != nullptr) {
        const float sc = invdeg[row];
        const float* arow = Am + (size_t)row * K;
        for (int k0 = 0; k0 < K; k0 += 4) {
            v2f a;                                // A frag: k = k0 + 2*half + v
            a.x = arow[k0 + 2 * half + 0] * sc;
            a.y = arow[k0 + 2 * half + 1] * sc;
#pragma unroll
            for (int t = 0; t < NT; ++t) {
                v2f b;                            // B frag: same k, n = t*16+l16
                b.x = Wm[(size_t)(k0 + 2 * half + 0) * ncol + t * 16 + l16];
                b.y = Wm[(size_t)(k0 + 2 * half + 1) * ncol + t * 16 + l16];
                acc[t] = __builtin_amdgcn_wmma_f32_16x16x4_f32(
                    false, a, false, b, (short)0, acc[t], false, false);
            }
        }
    }

    // ---- path 2: root features ----
    {
        const float* arow = Ax + (size_t)row * K;
        for (int k0 = 0; k0 < K; k0 += 4) {
            v2f a;
            a.x = arow[k0 + 2 * half + 0];
            a.y = arow[k0 + 2 * half + 1];
#pragma unroll
            for (int t = 0; t < NT; ++t) {
                v2f b;
                b.x = Wx[(size_t)(k0 + 2 * half + 0) * ncol + t * 16 + l16];
                b.y = Wx[(size_t)(k0 + 2 * half + 1) * ncol + t * 16 + l16];
                acc[t] = __builtin_amdgcn_wmma_f32_16x16x4_f32(
                    false, a, false, b, (short)0, acc[t], false, false);
            }
        }
    }

    // ---- epilogue: bias, store, BN column statistics ----
#pragma unroll
    for (int t = 0; t < NT; ++t) {
        const float bv = bias[t * 16 + l16];     // every lane value shares col n
#pragma unroll
        for (int r = 0; r < 8; ++r) acc[t][r] += bv;

#pragma unroll
        for (int r = 0; r < 8; ++r) {            // D layout: m = r + 8*half
            int grow = blockIdx.x * 16 + r + 8 * half;
            D[(size_t)grow * ncol + t * 16 + l16] = acc[t][r];
        }

        if (colsum != nullptr) {
            float s1 = 0.0f, s2 = 0.0f;
#pragma unroll
            for (int r = 0; r < 8; ++r) {
                float v = acc[t][r];
                s1 += v;
                s2 += v * v;
            }
            // lanes L and L^16 hold the same column (different row halves)
            s1 += __shfl_xor(s1, 16, 32);
            s2 += __shfl_xor(s2, 16, 32);
            if (half == 0) {
                atomicAdd(&colsum[t * 16 + l16], s1);
                atomicAdd(&colsumsq[t * 16 + l16], s2);
            }
        }
    }
}

// -------------------------------------------------- BatchNorm(+ReLU) apply
__global__ void bn_relu_kernel(const float* __restrict__ hpre, const float* __restrict__ stats,
                               const float* __restrict__ gamma, const float* __restrict__ beta,
                               float* __restrict__ out, int total) {
    int i = blockIdx.x * blockDim.x + threadIdx.x;
    if (i >= total) return;
    int col    = i & (DH - 1);
    float mu   = stats[col] * (1.0f / (float)NN);
    float var  = stats[DH + col] * (1.0f / (float)NN) - mu * mu;
    float g    = gamma[col] * rsqrtf(var + BN_EPS);
    float v    = (hpre[i] - mu) * g + beta[col];
    out[i] = fmaxf(v, 0.0f);
}

// ---------------------------------------------------------------- launcher
extern "C" void kernel_launch(void* const* d_in, const int* in_sizes, int n_in,
                              void* d_out, int out_size, void* d_ws, size_t ws_size,
                              hipStream_t stream) {
    (void)in_sizes; (void)n_in; (void)out_size; (void)ws_size;

    const float* x        = (const float*)d_in[0];
    const int*   e_sp     = (const int*)d_in[1];   // [2, E]: row0=src, row1=dst
    const int*   e_tm     = (const int*)d_in[2];
    const float* Wl[4]    = {(const float*)d_in[3],  (const float*)d_in[8],
                             (const float*)d_in[13], (const float*)d_in[18]};
    const float* Wr[4]    = {(const float*)d_in[4],  (const float*)d_in[9],
                             (const float*)d_in[14], (const float*)d_in[19]};
    const float* bb[4]    = {(const float*)d_in[5],  (const float*)d_in[10],
                             (const float*)d_in[15], (const float*)d_in[20]};
    const float* gamma[4] = {(const float*)d_in[6],  (const float*)d_in[11],
                             (const float*)d_in[16], (const float*)d_in[21]};
    const float* beta[4]  = {(const float*)d_in[7],  (const float*)d_in[12],
                             (const float*)d_in[17], (const float*)d_in[22]};
    const float* Wout     = (const float*)d_in[23];
    const float* bout     = (const float*)d_in[24];
    float* out            = (float*)d_out;

    // workspace carve (floats): s | stats(256) | hpre | h | deg(2N)
    float* ws    = (float*)d_ws;
    float* s     = ws;                        // N*DH
    float* stats = s + (size_t)NN * DH;       // 256 (colsum | colsumsq)
    float* hpre  = stats + 256;               // N*DH
    float* h     = hpre + (size_t)NN * DH;    // N*DH
    float* deg   = h + (size_t)NN * DH;       // 2*N (spatial | temporal), holds 1/max(deg,1)

    const int TB = 256;
    const int nfeat   = NN * DH;              // 12.8M
    const int scatter_threads = EDG * 32;

    // degrees (per edge type) -> reciprocal of clamped degree
    zero_kernel<<<(2 * NN + TB - 1) / TB, TB, 0, stream>>>(deg, 2 * NN);
    deg_count_kernel<<<(EDG + TB - 1) / TB, TB, 0, stream>>>(e_sp + EDG, deg,      EDG);
    deg_count_kernel<<<(EDG + TB - 1) / TB, TB, 0, stream>>>(e_tm + EDG, deg + NN, EDG);
    inv_deg_kernel<<<(2 * NN + TB - 1) / TB, TB, 0, stream>>>(deg, 2 * NN);

    for (int layer = 0; layer < 4; ++layer) {
        const int*   edges = (layer < 2) ? e_sp : e_tm;
        const float* invd  = (layer < 2) ? deg  : deg + NN;
        const float* xin   = (layer == 0) ? x : h;

        // zero segment-sum buffer + BN stat accumulators (contiguous)
        zero_kernel<<<(nfeat + 256 + TB - 1) / TB, TB, 0, stream>>>(s, nfeat + 256);

        scatter_kernel<<<(scatter_threads + TB - 1) / TB, TB, 0, stream>>>(
            edges, edges + EDG, xin, s, EDG);

        gemm_bn_kernel<DH / 16><<<NN / 16, 32, 0, stream>>>(
            s, invd, xin, Wl[layer], Wr[layer], bb[layer],
            hpre, stats, stats + DH, DH);

        bn_relu_kernel<<<(nfeat + TB - 1) / TB, TB, 0, stream>>>(
            hpre, stats, gamma[layer], beta[layer], h, nfeat);
    }

    // final projection: out = h @ Wout + bout   (NT = 32/16 = 2 col tiles)
    gemm_bn_kernel<DOUTC / 16><<<NN / 16, 32, 0, stream>>>(
        nullptr, nullptr, h, nullptr, Wout, bout, out, nullptr, nullptr, DH);
}